// Word2VecNegativeSamples_48447231099312
// MI455X (gfx1250) — compile-verified
//
#include <hip/hip_runtime.h>

// Word2Vec negative-sampling forward: out[b,k] = dot(W_out[out_idx[b,k],:], W_in[:,in_idx[b]])
// B=16384, K=21, EMBED=10. Memory-bound gather; reduction done on the fp32 WMMA pipe
// (v_wmma_f32_16x16x4_f32) with exact fp32 accumulation.

typedef float v2f __attribute__((ext_vector_type(2)));
typedef float v8f __attribute__((ext_vector_type(8)));

#define NUM_TOKENS 1000000
#define EMBED 10
#define KNEG 21

__global__ __launch_bounds__(256)
void w2v_neg_wmma_kernel(const int* __restrict__ in_idx,      // (B,)
                         const int* __restrict__ out_idx,     // (B, K)
                         const float* __restrict__ W_in,      // (EMBED, NUM_TOKENS) row-major
                         const float* __restrict__ W_out,     // (NUM_TOKENS, EMBED) row-major
                         float* __restrict__ out,             // (B, K)
                         int batch)
{
    const int lane  = threadIdx.x & 31;
    const int gwave = blockIdx.x * 8 + (threadIdx.x >> 5);   // 8 wave32 per 256-thread block
    const int b     = gwave >> 1;                            // batch element
    const int tile  = gwave & 1;                             // k rows [0,16) or [16,21)
    if (b >= batch) return;                                  // grid sized exactly; kept for safety

    const int h = lane >> 4;                                 // lane half (selects K pair)
    const int m = lane & 15;                                 // matrix row within tile

    // Clamp invalid rows of tile 1 (k = 21..31) onto k = 20; their outputs are discarded.
    int k = tile * 16 + m;
    if (k > KNEG - 1) k = KNEG - 1;

    const long long tok_out = out_idx[(long long)b * KNEG + k];
    const float* ovrow = W_out + tok_out * EMBED;            // contiguous 40B row

    const long long tok_in = in_idx[b];
    const float* ivcol = W_in + tok_in;                      // column: stride NUM_TOKENS floats

    // D = A x B + C accumulated over 3 K-chunks (EMBED=10 padded to 12).
    // A: 16x4 fp32; lane m+16h supplies A[m][d0+2h], A[m][d0+2h+1] in (a.x, a.y).
    // B: 4x16 fp32 with all 16 columns equal to iv[b]; lane supplies the same two
    //    K-indices (d0+2h, d0+2h+1) as its A slots.
    v8f c = {};
    #pragma unroll
    for (int chunk = 0; chunk < 3; ++chunk) {
        const int d0 = chunk * 4 + 2 * h;                    // first K index for this lane
        v2f a  = (v2f){0.f, 0.f};
        v2f bm = (v2f){0.f, 0.f};
        if (d0 + 1 < EMBED) {                                // pairs (0,1)..(8,9) valid; (10,11) zero
            a.x  = ovrow[d0];
            a.y  = ovrow[d0 + 1];
            bm.x = ivcol[(long long)d0 * NUM_TOKENS];
            bm.y = ivcol[(long long)(d0 + 1) * NUM_TOKENS];
        }
        // 8 args: (neg_a, A, neg_b, B, c_mod, C, reuse_a, reuse_b)
        c = __builtin_amdgcn_wmma_f32_16x16x4_f32(
                false, a, false, bm, (short)0, c, false, false);
    }

    // Every column of D is identical (broadcast B). Per the 16x16 f32 C/D layout:
    //   rows 0..7  -> lanes 0..15, VGPR r
    //   rows 8..15 -> lanes 16..31, VGPR r
    // So lane 0 holds out rows 0..7 in c[0..7]; lane 16 holds rows 8..15.
    const int rbase = (lane == 0) ? 0 : (lane == 16) ? 8 : -1;
    if (rbase >= 0) {
        #pragma unroll
        for (int r = 0; r < 8; ++r) {
            const int kk = tile * 16 + rbase + r;
            if (kk < KNEG)
                out[(long long)b * KNEG + kk] = c[r];
        }
    }
}

extern "C" void kernel_launch(void* const* d_in, const int* in_sizes, int n_in,
                              void* d_out, int out_size, void* d_ws, size_t ws_size,
                              hipStream_t stream) {
    (void)n_in; (void)out_size; (void)d_ws; (void)ws_size;
    const int*   in_idx  = (const int*)  d_in[0];  // (B,)
    const int*   out_idx = (const int*)  d_in[1];  // (B, K)
    const float* W_in    = (const float*)d_in[2];  // (EMBED, NUM_TOKENS)
    const float* W_out   = (const float*)d_in[3];  // (NUM_TOKENS, EMBED)
    float*       out     = (float*)d_out;          // (B, K)

    const int batch = in_sizes[0];                 // 16384
    // 2 wave-tiles per batch element, 8 waves per 256-thread block.
    const int total_waves = batch * 2;
    const int blocks = (total_waves + 7) / 8;

    w2v_neg_wmma_kernel<<<blocks, 256, 0, stream>>>(in_idx, out_idx, W_in, W_out, out, batch);
}